// ReducedNet_33200097198824
// MI455X (gfx1250) — compile-verified
//
#include <hip/hip_runtime.h>
#include <math.h>

typedef __attribute__((ext_vector_type(2))) float v2f;
typedef __attribute__((ext_vector_type(8))) float v8f;
typedef __attribute__((ext_vector_type(4))) int v4i;

#define QMAX 7.0f
#define BATCH 2048
#define IMG_D 31
#define IMG_H 16
#define IMG_W 31
#define IMG_ELEMS (IMG_D * IMG_H * IMG_W)   // 15376 floats = 61.5 KB, fits LDS
#define KD 5
#define KH 3
#define KW 5
#define CONV_TAPS (KD * KH * KW)            // 75
#define POOL_OUT 343                        // 7*7*7
#define KPAD 344                            // 343 padded to multiple of 4 (zero pad)
#define NFC1 128
#define NFC2 4

// workspace layout (float offsets)
#define OFF_QWC 0                                   // 80 (75 used)
#define OFF_BQ1 80                                  // (KPAD/2)*NFC1*2 = 44032
#define OFF_QW2 (OFF_BQ1 + (KPAD / 2) * NFC1 * 2)   // 512
#define OFF_F   (OFF_QW2 + 512)                     // BATCH*KPAD = 704512
#define OFF_A1  (OFF_F + BATCH * KPAD)              // BATCH*NFC1 = 262144
// total ~1.01M floats ~= 3.9 MB

// gfx1250 async memory->LDS path (ASYNCcnt-tracked, bypasses VGPRs).
// Guarded so the file still compiles on toolchains without these builtins
// (and during the host pass, where amdgcn builtins are not visible).
#if defined(__HIP_DEVICE_COMPILE__) &&                         \
    __has_builtin(__builtin_amdgcn_global_load_async_to_lds_b128) && \
    __has_builtin(__builtin_amdgcn_s_wait_asynccnt)
#define USE_ASYNC_LDS 1
#endif

#ifdef USE_ASYNC_LDS
// Builtin signature (from hipcc diagnostic): (int4 AS1* src, int4 AS3* dst,
// imm int offset, imm int cpol). AS1 prints as __device__ in HIP.
typedef __attribute__((address_space(1))) v4i gv4i;  // global int4
typedef __attribute__((address_space(3))) v4i lv4i;  // LDS int4
#endif

// ---------------------------------------------------------------------------
// Kernel 0: per-tensor symmetric 4-bit fake-quant (forward value = dequant q).
// block 0: w_conv (75) -> ws[OFF_QWC]
// block 1: w_fc1 (128x343) -> transposed, K-paired layout ws[OFF_BQ1]:
//          bq1[(k>>1)*256 + n*2 + (k&1)]  (so (k,k+1) for a column is a float2)
// block 2: w_fc2 (4x128) -> ws[OFF_QW2]
// ---------------------------------------------------------------------------
__global__ __launch_bounds__(256) void quant_kernel(
    const float* __restrict__ wc, const float* __restrict__ w1,
    const float* __restrict__ w2, float* __restrict__ ws) {
  __shared__ float red[256];
  const int t = threadIdx.x;
  const int which = blockIdx.x;
  const float* src;
  int n;
  if (which == 0)      { src = wc; n = CONV_TAPS; }
  else if (which == 1) { src = w1; n = NFC1 * 343; }
  else                 { src = w2; n = NFC2 * NFC1; }

  float m = 0.f;
  for (int i = t; i < n; i += 256) m = fmaxf(m, fabsf(src[i]));
  red[t] = m;
  __syncthreads();
  for (int s = 128; s > 0; s >>= 1) {
    if (t < s) red[t] = fmaxf(red[t], red[t + s]);
    __syncthreads();
  }
  const float s = red[0] / QMAX;
  const float inv = (s > 0.f) ? (1.f / s) : 0.f;

  if (which == 0) {
    float* dst = ws + OFF_QWC;
    for (int i = t; i < n; i += 256)
      dst[i] = fminf(fmaxf(rintf(src[i] * inv), -8.f), 7.f) * s;
  } else if (which == 1) {
    float* dst = ws + OFF_BQ1;
    for (int i = t; i < n; i += 256) {
      const int o = i / 343, k = i % 343;
      const float q = fminf(fmaxf(rintf(src[i] * inv), -8.f), 7.f) * s;
      dst[(k >> 1) * (2 * NFC1) + o * 2 + (k & 1)] = q;
    }
    // zero-fill the K=343 pad slot (odd slot of pair index 171)
    for (int o = t; o < NFC1; o += 256)
      dst[(343 >> 1) * (2 * NFC1) + o * 2 + 1] = 0.f;
  } else {
    float* dst = ws + OFF_QW2;
    for (int i = t; i < n; i += 256)
      dst[i] = fminf(fmaxf(rintf(src[i] * inv), -8.f), 7.f) * s;
  }
}

// ---------------------------------------------------------------------------
// Kernel 1: fused conv3d(stride 2,1,2, VALID) + bias + maxpool3d(2).
// One block per batch image. Image staged in LDS with the gfx1250 async
// memory->LDS path (ASYNCcnt) when available, else float4 loads via VGPRs.
// Writes f[b][0..342] (+ zero pad at col 343) for the WMMA GEMM.
// ---------------------------------------------------------------------------
__global__ __launch_bounds__(256) void convpool_kernel(
    const float* __restrict__ x, const float* __restrict__ bconv,
    const float* __restrict__ qwc, float* __restrict__ f) {
  __shared__ float4 simg4[IMG_ELEMS / 4];   // 15376/4 = 3844
  __shared__ float swt[CONV_TAPS + 1];
  float* simg = (float*)simg4;

  const int t = threadIdx.x;
  const int b = blockIdx.x;

  const float4* src = (const float4*)(x + (size_t)b * IMG_ELEMS);
#ifdef USE_ASYNC_LDS
  // global_load_async_to_lds_b128: per-lane 16B memory->LDS, no VGPR bounce.
  for (int i = t; i < IMG_ELEMS / 4; i += 256)
    __builtin_amdgcn_global_load_async_to_lds_b128(
        (gv4i*)(src + i), (lv4i*)(simg4 + i), 0, 0);
  if (t < CONV_TAPS) swt[t] = qwc[t];
  __builtin_amdgcn_s_wait_asynccnt(0);  // this wave's async loads landed
  __syncthreads();                      // => every wave's loads landed
#else
  for (int i = t; i < IMG_ELEMS / 4; i += 256) simg4[i] = src[i];
  if (t < CONV_TAPS) swt[t] = qwc[t];
  __syncthreads();
#endif

  const float bias = bconv[0];
  for (int po = t; po < POOL_OUT; po += 256) {
    const int pd = po / 49;
    const int rem = po % 49;
    const int ph = rem / 7;
    const int pw = rem % 7;
    float best = -INFINITY;
    #pragma unroll
    for (int dd = 0; dd < 2; ++dd) {
      #pragma unroll
      for (int hh = 0; hh < 2; ++hh) {
        #pragma unroll
        for (int wp = 0; wp < 2; ++wp) {
          const int d0 = 2 * (2 * pd + dd);   // input D start (stride 2)
          const int h0 = (2 * ph + hh);       // input H start (stride 1)
          const int w0 = 2 * (2 * pw + wp);   // input W start (stride 2)
          float acc = 0.f;
          #pragma unroll
          for (int kd = 0; kd < KD; ++kd) {
            #pragma unroll
            for (int kh = 0; kh < KH; ++kh) {
              const float* row =
                  &simg[(d0 + kd) * (IMG_H * IMG_W) + (h0 + kh) * IMG_W + w0];
              const float* w = &swt[kd * (KH * KW) + kh * KW];
              #pragma unroll
              for (int kw = 0; kw < KW; ++kw) acc = fmaf(row[kw], w[kw], acc);
            }
          }
          best = fmaxf(best, acc);
        }
      }
    }
    f[(size_t)b * KPAD + po] = best + bias;
  }
  if (t == 0) f[(size_t)b * KPAD + 343] = 0.f;
}

// ---------------------------------------------------------------------------
// Kernel 2: FC1 via V_WMMA_F32_16X16X4_F32 (full-f32 matrix path).
// C[2048,128] = f[2048,344] x Bq[344,128]; one wave per 16x16 tile.
// Fragment layouts per ISA 7.12.2:
//   A: lanes 0-15 rows m0+l with K={k0,k0+1}; lanes 16-31 K={k0+2,k0+3}
//   B: lanes 0-15 cols n0+l with K={k0,k0+1}; lanes 16-31 K={k0+2,k0+3}
//   C: VGPR r -> row m0 + r + 8*(lane>=16), col n0 + (lane&15)
// ---------------------------------------------------------------------------
__global__ __launch_bounds__(256) void fc1_wmma_kernel(
    const float* __restrict__ f, const float* __restrict__ bq1,
    const float* __restrict__ b1, float* __restrict__ a1) {
  const int lane = threadIdx.x & 31;
  const int wave = threadIdx.x >> 5;      // 8 waves -> N tiles
  const int m0 = blockIdx.x * 16;
  const int n0 = wave * 16;
  const int half = lane >> 4;             // 0: K pair {0,1}; 1: K pair {2,3}
  const int l = lane & 15;

  const float* aptr = f + (size_t)(m0 + l) * KPAD + half * 2;
  // B pair (k,k+1) for column `col` lives at bq1[(k>>1)*256 + col*2]
  const float* bptr = bq1 + (size_t)(n0 + l) * 2 + half * 256;

  v8f acc = {};
  for (int k0 = 0; k0 < KPAD; k0 += 4) {
    const v2f a = *(const v2f*)(aptr + k0);
    const v2f bb = *(const v2f*)(bptr + (k0 >> 1) * 256);
    acc = __builtin_amdgcn_wmma_f32_16x16x4_f32(
        false, a, false, bb, (short)0, acc, false, false);
  }

  #pragma unroll
  for (int r = 0; r < 8; ++r) {
    const int row = m0 + r + half * 8;
    const int col = n0 + l;
    const float v = acc[r] + b1[col];
    a1[(size_t)row * NFC1 + col] = fmaxf(v, 0.f);   // ReLU
  }
}

// ---------------------------------------------------------------------------
// Kernel 3: FC2 (128 -> 4) + softmax. One thread per batch row.
// ---------------------------------------------------------------------------
__global__ __launch_bounds__(256) void fc2_softmax_kernel(
    const float* __restrict__ a1, const float* __restrict__ qw2,
    const float* __restrict__ b2, float* __restrict__ out) {
  const int b = blockIdx.x * blockDim.x + threadIdx.x;
  if (b >= BATCH) return;
  const float* row = a1 + (size_t)b * NFC1;
  float s[NFC2];
  #pragma unroll
  for (int j = 0; j < NFC2; ++j) s[j] = b2[j];
  for (int i = 0; i < NFC1; ++i) {
    const float v = row[i];
    #pragma unroll
    for (int j = 0; j < NFC2; ++j) s[j] = fmaf(v, qw2[j * NFC1 + i], s[j]);
  }
  const float m = fmaxf(fmaxf(s[0], s[1]), fmaxf(s[2], s[3]));
  float e[NFC2], sum = 0.f;
  #pragma unroll
  for (int j = 0; j < NFC2; ++j) { e[j] = expf(s[j] - m); sum += e[j]; }
  const float inv = 1.f / sum;
  #pragma unroll
  for (int j = 0; j < NFC2; ++j) out[(size_t)b * NFC2 + j] = e[j] * inv;
}

extern "C" void kernel_launch(void* const* d_in, const int* in_sizes, int n_in,
                              void* d_out, int out_size, void* d_ws,
                              size_t ws_size, hipStream_t stream) {
  const float* x      = (const float*)d_in[0];
  const float* w_conv = (const float*)d_in[1];
  const float* b_conv = (const float*)d_in[2];
  const float* w_fc1  = (const float*)d_in[3];
  const float* b_fc1  = (const float*)d_in[4];
  const float* w_fc2  = (const float*)d_in[5];
  const float* b_fc2  = (const float*)d_in[6];
  float* ws  = (float*)d_ws;
  float* out = (float*)d_out;
  (void)in_sizes; (void)n_in; (void)out_size; (void)ws_size;

  quant_kernel<<<3, 256, 0, stream>>>(w_conv, w_fc1, w_fc2, ws);
  convpool_kernel<<<BATCH, 256, 0, stream>>>(x, b_conv, ws + OFF_QWC,
                                             ws + OFF_F);
  fc1_wmma_kernel<<<BATCH / 16, 256, 0, stream>>>(ws + OFF_F, ws + OFF_BQ1,
                                                  b_fc1, ws + OFF_A1);
  fc2_softmax_kernel<<<BATCH / 256, 256, 0, stream>>>(ws + OFF_A1,
                                                      ws + OFF_QW2, b_fc2, out);
}